// CropPool2D_73813307949638
// MI455X (gfx1250) — compile-verified
//
#include <hip/hip_runtime.h>

// CropPool2D for MI455X (gfx1250, wave32).
// B=64, C=512, H=W=56. out[b,c] = mean of img_feats[b,c, y1:y2, x1:x2].
//
// One wave per (batch, 16-channel group). Lane m (0-15) streams the even ROI
// rows of channel c0+m, lane m+16 the odd rows, each into 2 scalar
// accumulators (the lane's two A-matrix K-slots). A single
// V_WMMA_F32_16X16X4_F32 with B = all-ones then performs the entire
// cross-lane reduction: D[m][n] = sum of channel m's 4 partials, deposited
// per the documented C/D layout (lane 0 -> M=0..7, lane 16 -> M=8..15).
// B=ones makes the result independent of the undocumented B VGPR layout.

#define B_N 64
#define C_N 512
#define H_N 56
#define W_N 56

typedef float v2f __attribute__((ext_vector_type(2)));
typedef float v8f __attribute__((ext_vector_type(8)));

__global__ __launch_bounds__(32)
void croppool_wmma_kernel(const float* __restrict__ feats,
                          const int*   __restrict__ bboxes,
                          float*       __restrict__ out) {
  const int b    = blockIdx.y;
  const int c0   = blockIdx.x * 16;        // 16-channel group
  const int lane = threadIdx.x;            // 0..31 (wave32)
  const int m    = lane & 15;              // channel within group
  const int h    = lane >> 4;              // row-parity split between lane halves

  // bbox is per-batch -> wave-uniform bounds
  const int x1 = bboxes[b * 4 + 0];
  const int y1 = bboxes[b * 4 + 1];
  const int x2 = bboxes[b * 4 + 2];
  const int y2 = bboxes[b * 4 + 3];

  const float inv_area = 1.0f / (float)((x2 - x1) * (y2 - y1));

  const float* base = feats + (size_t)(b * C_N + c0 + m) * (H_N * W_N);

  // Per-lane partial sums: this lane's two A-matrix K-slots.
  float s0 = 0.0f, s1 = 0.0f;

  for (int y = y1 + h; y < y2; y += 2) {
    const float* row = base + y * W_N;
    if (y + 2 < y2) {
      __builtin_prefetch(base + (y + 2) * W_N + x1, 0, 1);  // global_prefetch_b8
    }
    int x = x1;
    // Unrolled-by-4 streaming over the contiguous ROI row segment.
    for (; x + 3 < x2; x += 4) {
      s0 += row[x]     + row[x + 2];
      s1 += row[x + 1] + row[x + 3];
    }
    for (; x < x2; ++x) s0 += row[x];
  }

  // Single WMMA = full cross-lane reduction. A[m][0..1] from lane m,
  // A[m][2..3] from lane m+16 (documented 16x4 f32 A layout); B = ones.
  v2f a;
  a[0] = s0;
  a[1] = s1;
  v2f bones;
  bones[0] = 1.0f;
  bones[1] = 1.0f;
  v8f acc = {};
  acc = __builtin_amdgcn_wmma_f32_16x16x4_f32(
      /*neg_a=*/false, a, /*neg_b=*/false, bones,
      /*c_mod=*/(short)0, acc, /*reuse_a=*/false, /*reuse_b=*/false);

  // C/D layout: lane 0 holds M=0..7 in VGPRs 0..7, lane 16 holds M=8..15.
  if (lane == 0) {
    float* o = out + (size_t)b * C_N + c0;
#pragma unroll
    for (int r = 0; r < 8; ++r) o[r] = acc[r] * inv_area;
  } else if (lane == 16) {
    float* o = out + (size_t)b * C_N + c0 + 8;
#pragma unroll
    for (int r = 0; r < 8; ++r) o[r] = acc[r] * inv_area;
  }
}

extern "C" void kernel_launch(void* const* d_in, const int* in_sizes, int n_in,
                              void* d_out, int out_size, void* d_ws, size_t ws_size,
                              hipStream_t stream) {
  const float* feats  = (const float*)d_in[0];   // (64, 512, 56, 56) f32
  const int*   bboxes = (const int*)d_in[1];     // (64, 4) i32
  float*       out    = (float*)d_out;           // (64, 512) f32

  dim3 grid(C_N / 16, B_N);   // 32 channel groups x 64 batches = 2048 waves
  dim3 block(32);             // one wave32 per block
  croppool_wmma_kernel<<<grid, block, 0, stream>>>(feats, bboxes, out);
}